// MoEGate_41936060678342
// MI455X (gfx1250) — compile-verified
//
#include <hip/hip_runtime.h>
#include <math.h>

typedef __attribute__((ext_vector_type(2))) float v2f;
typedef __attribute__((ext_vector_type(4))) float v4f;
typedef __attribute__((ext_vector_type(8))) float v8f;

#define B_DIM 4
#define S_DIM 4096
#define D_DIM 2048
#define E_DIM 64
#define K_DIM 2
#define NTOK  (B_DIM * S_DIM)                  /* 16384 */
#define DISP_ELEMS ((size_t)NTOK * E_DIM * K_DIM) /* 2097152 */

// ---------------------------------------------------------------- zero scratch
__global__ void moe_zero_ws(float* __restrict__ ws) {
    int i = threadIdx.x;
    if (i < 129) ws[i] = 0.0f;   // [0,64) gate_sum, [64,128) sel_cnt, [128] z_sum
}

// ---------------------------------------------------------------- main kernel
// One wave32 computes a 16-token x 64-expert tile with fp32 WMMA, then does
// softmax / top-2 / output scatter / loss accumulation for its 16 tokens.
__global__ __launch_bounds__(256) void moe_gate_kernel(
    const float* __restrict__ x,   // [NTOK, D]
    const float* __restrict__ w,   // [E, D]
    float* __restrict__ out,       // dispatch | combine | 2 losses
    float* __restrict__ ws)        // 129 floats of accumulators
{
    __shared__ float lds_log[8][16 * 66];  // per-wave logits, padded stride 66
    __shared__ float lds_res[8][64];       // per-wave {i0,i1,p0,p1} x 16 tokens

    const int lane = threadIdx.x & 31;
    const int wib  = threadIdx.x >> 5;          // wave index in block (0..7)
    const int wave = blockIdx.x * 8 + wib;      // global wave id
    const int row0 = wave * 16;                 // first token of this tile
    const int half = lane >> 4;                 // 0: lanes 0-15, 1: lanes 16-31
    const int m    = lane & 15;

    // ---- GEMM: logits[16 x 64] = x_tile @ W^T via V_WMMA_F32_16X16X4_F32 ----
    // A (16x4 fp32): lane holds M=m, {K=2*half, K=2*half+1}  -> float2 load
    // B (4x16 fp32): lane holds N=m, {K=2*half, K=2*half+1}  -> float2 load
    const float* xrow = x + (size_t)(row0 + m) * D_DIM + 2 * half;
    const float* wr0  = w + (size_t)( 0 + m) * D_DIM + 2 * half;
    const float* wr1  = w + (size_t)(16 + m) * D_DIM + 2 * half;
    const float* wr2  = w + (size_t)(32 + m) * D_DIM + 2 * half;
    const float* wr3  = w + (size_t)(48 + m) * D_DIM + 2 * half;

    v8f acc0 = {}; v8f acc1 = {}; v8f acc2 = {}; v8f acc3 = {};

#pragma unroll 4
    for (int k = 0; k < D_DIM; k += 4) {
        v2f a  = *(const v2f*)(xrow + k);
        v2f b0 = *(const v2f*)(wr0 + k);
        v2f b1 = *(const v2f*)(wr1 + k);
        v2f b2 = *(const v2f*)(wr2 + k);
        v2f b3 = *(const v2f*)(wr3 + k);
        acc0 = __builtin_amdgcn_wmma_f32_16x16x4_f32(false, a, false, b0,
                                                     (short)0, acc0, false, false);
        acc1 = __builtin_amdgcn_wmma_f32_16x16x4_f32(false, a, false, b1,
                                                     (short)0, acc1, false, false);
        acc2 = __builtin_amdgcn_wmma_f32_16x16x4_f32(false, a, false, b2,
                                                     (short)0, acc2, false, false);
        acc3 = __builtin_amdgcn_wmma_f32_16x16x4_f32(false, a, false, b3,
                                                     (short)0, acc3, false, false);
    }

    // ---- dump C tiles to LDS (row = token, padded stride 66: conflict-free) ----
    // C layout: acc[r] @ lane = logits[M = r + 8*half, N = lane%16]
    float* L = lds_log[wib];
#pragma unroll
    for (int r = 0; r < 8; ++r) {
        int mm = r + 8 * half;
        L[mm * 66 +  0 + m] = acc0[r];
        L[mm * 66 + 16 + m] = acc1[r];
        L[mm * 66 + 32 + m] = acc2[r];
        L[mm * 66 + 48 + m] = acc3[r];
    }
    __syncthreads();

    // ---- per-token softmax / top-2 / losses: lanes 0..15, one token each ----
    if (lane < 16) {
        const float* row = L + lane * 66;
        float mx = row[0];
#pragma unroll
        for (int i = 1; i < 64; ++i) mx = fmaxf(mx, row[i]);

        float sum = 0.0f;
        float v0 = -INFINITY, v1 = -INFINITY;
        int   i0 = 0, i1 = 0;
#pragma unroll
        for (int i = 0; i < 64; ++i) {
            float l = row[i];
            sum += __expf(l - mx);
            if (l > v0)      { v1 = v0; i1 = i0; v0 = l; i0 = i; }
            else if (l > v1) { v1 = l;  i1 = i; }
        }
        float inv = 1.0f / sum;
        float p0 = __expf(v0 - mx) * inv;
        float p1 = __expf(v1 - mx) * inv;

        // router z-loss quirk: logsumexp over the *probabilities*
        float zs = 0.0f;
#pragma unroll
        for (int i = 0; i < 64; ++i) zs += __expf(__expf(row[i] - mx) * inv);
        float lse = __logf(zs);

        atomicAdd(&ws[i0], p0);            // gate_sum
        atomicAdd(&ws[i1], p1);
        atomicAdd(&ws[64 + i0], 1.0f);     // selection count
        atomicAdd(&ws[64 + i1], 1.0f);
        atomicAdd(&ws[128], lse * lse);    // z-loss accumulator

        float* R = lds_res[wib] + lane * 4;
        R[0] = __int_as_float(i0);
        R[1] = __int_as_float(i1);
        R[2] = p0;
        R[3] = p1;
    }
    __syncthreads();

    // ---- coalesced output: full [E,K]=128-float rows, zeros included ----
    float* disp = out;
    float* comb = out + DISP_ELEMS;
    const float* R = lds_res[wib];
#pragma unroll
    for (int t = 0; t < 16; ++t) {
        int   i0 = __float_as_int(R[t * 4 + 0]);
        int   i1 = __float_as_int(R[t * 4 + 1]);
        float p0 = R[t * 4 + 2];
        float p1 = R[t * 4 + 3];
        size_t base = (size_t)(row0 + t) * (E_DIM * K_DIM) + lane * 4;
        int e0 = lane * 2, e1 = lane * 2 + 1;
        v4f dv = { (e0 == i0 || e0 == i1) ? 1.0f : 0.0f, 0.0f,
                   (e1 == i0 || e1 == i1) ? 1.0f : 0.0f, 0.0f };
        v4f cv = { e0 == i0 ? p0 : (e0 == i1 ? p1 : 0.0f), 0.0f,
                   e1 == i0 ? p0 : (e1 == i1 ? p1 : 0.0f), 0.0f };
        *(v4f*)(disp + base) = dv;
        *(v4f*)(comb + base) = cv;
    }
}

// ---------------------------------------------------------------- finalize
__global__ void moe_finalize(const float* __restrict__ ws, float* __restrict__ out) {
    if (threadIdx.x == 0) {
        const float invn = 1.0f / (float)NTOK;
        float lbl = 0.0f;
        for (int e = 0; e < E_DIM; ++e)
            lbl += (ws[e] * invn) * (ws[64 + e] * invn);
        out[2 * DISP_ELEMS]     = lbl * (float)E_DIM;  // load_balancing_loss
        out[2 * DISP_ELEMS + 1] = ws[128] * invn;      // router_z_loss
    }
}

// ---------------------------------------------------------------- launcher
extern "C" void kernel_launch(void* const* d_in, const int* in_sizes, int n_in,
                              void* d_out, int out_size, void* d_ws, size_t ws_size,
                              hipStream_t stream) {
    const float* x  = (const float*)d_in[0];   // hidden_states [4,4096,2048] fp32
    const float* w  = (const float*)d_in[1];   // gate_weight   [64,2048]     fp32
    float* out = (float*)d_out;
    float* ws  = (float*)d_ws;

    moe_zero_ws<<<1, 256, 0, stream>>>(ws);
    // 16384 tokens / 16 per wave = 1024 waves; 8 waves (256 thr) per block.
    moe_gate_kernel<<<NTOK / 16 / 8, 256, 0, stream>>>(x, w, out, ws);
    moe_finalize<<<1, 64, 0, stream>>>(ws, out);
}